// TotalMatcher_62053687492846
// MI455X (gfx1250) — compile-verified
//
#include <hip/hip_runtime.h>
#include <math.h>

// ---------------------------------------------------------------------------
// TotalMatcher for MI455X (gfx1250, wave32).
// Memory-bound problem: spikes = 256MB streamed twice (stats pass + WMMA
// distance pass) -> ~22us HBM floor at 23.3 TB/s. GEMM part (1M x 64 x 32,
// 4.1 GFLOP f32) done with V_WMMA_F32_16X16X4_F32 so math is never the limit.
// ---------------------------------------------------------------------------

typedef float v2f __attribute__((ext_vector_type(2)));
typedef float v8f __attribute__((ext_vector_type(8)));

#define SDIM 64
#define KCL  32
// float-indexed workspace layout
#define WS_SUM   0        // K*S
#define WS_SQ    2048     // K*S
#define WS_CNT   4096     // K
#define WS_MEAN  4128     // K*S
#define WS_T2    6176     // K
#define WS_THR   6208     // K  (3*std, or -1 for invalid cluster -> always masked)

#define FACTOR_F 3.0f
#define MAXD_TIMES_S 1280.0f   // 20.0 * 64

// --- K0: zero the atomic-accumulation region (harness poisons ws with 0xAA) --
__global__ void tm_zero(float* __restrict__ ws) {
  int i = blockIdx.x * blockDim.x + threadIdx.x;
  if (i < WS_MEAN) ws[i] = 0.0f;
}

// --- K1: segment sums via LDS ds_add_f32, one spike per wave-iteration ------
// LDS layout [c*64+f]: wave lanes add features 2*lane,2*lane+1 of ONE spike
// (cluster uniform across wave) -> 32 distinct banks per ds_add, no conflicts.
__global__ void __launch_bounds__(256) tm_accum(
    const float* __restrict__ spikes, const int* __restrict__ sort_idx,
    float* __restrict__ ws, int N) {
  __shared__ float lbuf[WS_MEAN];           // 4128 floats = 16.5 KB
  for (int i = threadIdx.x; i < WS_MEAN; i += blockDim.x) lbuf[i] = 0.0f;
  __syncthreads();
  float* lsum = lbuf;
  float* lsq  = lbuf + WS_SQ;
  float* lcnt = lbuf + WS_CNT;

  const int lane = threadIdx.x & 31;
  const int wid  = (blockIdx.x * blockDim.x + threadIdx.x) >> 5;
  const int nw   = (gridDim.x * blockDim.x) >> 5;

  auto acc1 = [&](v2f d, int c) {
    int b = c * SDIM + 2 * lane;
    atomicAdd(&lsum[b],     d.x);
    atomicAdd(&lsum[b + 1], d.y);
    atomicAdd(&lsq[b],      d.x * d.x);
    atomicAdd(&lsq[b + 1],  d.y * d.y);
    if (lane == 0) atomicAdd(&lcnt[c], 1.0f);
  };

  int n = wid;
  // 4-spike batches: issue 4 independent 256B row loads before the LDS adds
  // so LOADcnt keeps several lines in flight per wave.
  while (n + 3 * nw < N) {
    int n0 = n, n1 = n + nw, n2 = n + 2 * nw, n3 = n + 3 * nw;
    v2f d0 = *(const v2f*)(spikes + (size_t)n0 * SDIM + 2 * lane);
    v2f d1 = *(const v2f*)(spikes + (size_t)n1 * SDIM + 2 * lane);
    v2f d2 = *(const v2f*)(spikes + (size_t)n2 * SDIM + 2 * lane);
    v2f d3 = *(const v2f*)(spikes + (size_t)n3 * SDIM + 2 * lane);
    int c0 = sort_idx[n0], c1 = sort_idx[n1], c2 = sort_idx[n2], c3 = sort_idx[n3];
    acc1(d0, c0); acc1(d1, c1); acc1(d2, c2); acc1(d3, c3);
    n += 4 * nw;
  }
  while (n < N) {
    v2f d = *(const v2f*)(spikes + (size_t)n * SDIM + 2 * lane);
    acc1(d, sort_idx[n]);
    n += nw;
  }
  __syncthreads();
  for (int i = threadIdx.x; i < WS_MEAN; i += blockDim.x)
    atomicAdd(&ws[i], lbuf[i]);
}

// --- K2: one wave per cluster -> means, t2 = ||mean||^2, thr = 3*std --------
__global__ void tm_finalize(float* __restrict__ ws) {
  int c    = threadIdx.x >> 5;   // cluster 0..31
  int lane = threadIdx.x & 31;
  float cnt  = ws[WS_CNT + c];
  float safe = fmaxf(cnt, 1.0f);
  int b = c * SDIM + lane;
  float s0 = ws[WS_SUM + b], s1 = ws[WS_SUM + b + 32];
  float q0 = ws[WS_SQ + b],  q1 = ws[WS_SQ + b + 32];
  float m0 = s0 / safe, m1 = s1 / safe;
  ws[WS_MEAN + b]      = m0;
  ws[WS_MEAN + b + 32] = m1;
  float var = (q0 / safe - m0 * m0) + (q1 / safe - m1 * m1);
  float t2  = m0 * m0 + m1 * m1;
  for (int m = 16; m >= 1; m >>= 1) {
    var += __shfl_xor(var, m, 32);
    t2  += __shfl_xor(t2,  m, 32);
  }
  if (lane == 0) {
    bool valid = (c != 0) && (cnt > 0.0f);     // CLID_UNMATCHED==0 excluded
    ws[WS_T2 + c]  = t2;
    ws[WS_THR + c] = valid ? FACTOR_F * sqrtf(fmaxf(var, 0.0f)) : -1.0f;
  }
}

// --- K3: WMMA distance pass. One wave owns a 16-spike tile ------------------
// A (16x4 f32, ISA layout): lane M=l&15; VGPR0/1 hold K=2*(l>>4)+{0,1}.
// B (4x16, mirrored):       lane N=l&15; VGPR0/1 hold K=2*(l>>4)+{0,1}.
// D (16x16 f32): VGPR r, lanes0-15: (M=r, N=lane); lanes16-31: (M=r+8, N=l-16).
__global__ void __launch_bounds__(256) tm_match(
    const float* __restrict__ spikes,
    const int*  __restrict__ sort_idx,
    const int*  __restrict__ match_idx,
    const float* __restrict__ distance,
    const float* __restrict__ ws,
    float* __restrict__ out, int N) {
  __shared__ v2f  bfrag[2][16][32];   // [col-tile][k-step][lane] = 8 KB
  __shared__ float t2s[KCL], thrs[KCL];

  // Stage B fragments into the exact per-lane WMMA layout (lane-stride 8B ->
  // conflict-free ds_load_b64 in the hot loop).
  for (int s = threadIdx.x; s < 1024; s += blockDim.x) {
    int t = s >> 9, k = (s >> 5) & 15, ln = s & 31;
    int c = t * 16 + (ln & 15);
    int f = 4 * k + 2 * (ln >> 4);
    v2f b;
    b.x = ws[WS_MEAN + c * SDIM + f];
    b.y = ws[WS_MEAN + c * SDIM + f + 1];
    bfrag[t][k][ln] = b;
  }
  if (threadIdx.x < KCL) {
    t2s[threadIdx.x]  = ws[WS_T2 + threadIdx.x];
    thrs[threadIdx.x] = ws[WS_THR + threadIdx.x];
  }
  __syncthreads();

  const int lane = threadIdx.x & 31;
  const int half = lane >> 4;
  const int l15  = lane & 15;
  const int wid  = (blockIdx.x * blockDim.x + threadIdx.x) >> 5;
  const int nw   = (gridDim.x * blockDim.x) >> 5;
  const int ntiles = (N + 15) >> 4;
  const float INF = __builtin_inff();

  for (int tile = wid; tile < ntiles; tile += nw) {
    int row  = tile * 16 + l15;
    int rowc = row < N ? row : N - 1;       // clamp tail loads (stores guarded)
    const v2f* arow = (const v2f*)(spikes + (size_t)rowc * SDIM + 2 * half);

    v8f acc_lo = {};
    v8f acc_hi = {};
    float x2p = 0.0f;
    #pragma unroll
    for (int k = 0; k < 16; ++k) {
      v2f a   = arow[2 * k];                // features 4k+2*half, +1
      x2p    += a.x * a.x + a.y * a.y;      // this lane's share of ||row||^2
      v2f blo = bfrag[0][k][lane];
      v2f bhi = bfrag[1][k][lane];
      acc_lo = __builtin_amdgcn_wmma_f32_16x16x4_f32(
          false, a, false, blo, (short)0, acc_lo, false, false);
      acc_hi = __builtin_amdgcn_wmma_f32_16x16x4_f32(
          false, a, false, bhi, (short)0, acc_hi, false, false);
    }
    // lane L + lane L^16 together cover row (L&15): combine halves.
    float x2full = x2p + __shfl_xor(x2p, 16, 32);

    float t2lo = t2s[l15],  t2hi = t2s[l15 + 16];
    float thrlo = thrs[l15], thrhi = thrs[l15 + 16];

    float dres[8]; int ires[8];
    #pragma unroll
    for (int r = 0; r < 8; ++r) {
      // row for this VGPR: m = r + 8*half; fetch its ||x||^2 (held by lane m).
      float x2r  = __shfl(x2full, (half << 3) + r, 32);
      float d2lo = x2r - 2.0f * acc_lo[r] + t2lo;
      float d2hi = x2r - 2.0f * acc_hi[r] + t2hi;
      float dlo  = sqrtf(fmaxf(d2lo, 1e-12f));
      float dhi  = sqrtf(fmaxf(d2hi, 1e-12f));
      dlo = (dlo > thrlo) ? INF : dlo;      // invalid cluster: thr=-1 -> INF
      dhi = (dhi > thrhi) ? INF : dhi;
      float bd; int bi;
      if (dhi < dlo) { bd = dhi; bi = l15 + 16; } else { bd = dlo; bi = l15; }
      #pragma unroll
      for (int m = 1; m <= 8; m <<= 1) {    // reduce over N within 16-lane half
        float od = __shfl_xor(bd, m, 32);
        int   oi = __shfl_xor(bi, m, 32);
        if (od < bd || (od == bd && oi < bi)) { bd = od; bi = oi; }
      }
      dres[r] = bd; ires[r] = bi;           // uniform within each half
    }

    // Re-gather so lane L (<16) holds row m=L: pick r=L&7, hi rows from L+16.
    int rsel = lane & 7;
    float dsel = dres[0]; int isel = ires[0];
    #pragma unroll
    for (int r = 1; r < 8; ++r)
      if (rsel == r) { dsel = dres[r]; isel = ires[r]; }
    int src = (lane & 8) ? (lane + 16) : lane;
    float dmin = __shfl(dsel, src, 32);
    int   imin = __shfl(isel, src, 32);

    if (lane < 16) {
      int spike = tile * 16 + lane;
      if (spike < N) {
        int  oldc = sort_idx[spike];
        int  cid  = (dmin >= MAXD_TIMES_S) ? 0 : imin;
        bool unm  = (oldc == 0);
        out[spike]                   = unm ? (float)cid : (float)oldc;
        out[(size_t)N + spike]       = unm ? 2.0f : (float)match_idx[spike];
        out[2 * (size_t)N + spike]   = unm ? dmin : distance[spike];
      }
    }
  }
}

extern "C" void kernel_launch(void* const* d_in, const int* in_sizes, int n_in,
                              void* d_out, int out_size, void* d_ws, size_t ws_size,
                              hipStream_t stream) {
  (void)n_in; (void)out_size; (void)ws_size;
  const float* spikes    = (const float*)d_in[0];
  const int*   sort_idx  = (const int*)d_in[1];
  const int*   match_idx = (const int*)d_in[2];
  const float* distance  = (const float*)d_in[3];
  const int    N         = in_sizes[1];
  float* ws  = (float*)d_ws;
  float* out = (float*)d_out;

  tm_zero<<<(WS_MEAN + 255) / 256, 256, 0, stream>>>(ws);
  tm_accum<<<256, 256, 0, stream>>>(spikes, sort_idx, ws, N);
  tm_finalize<<<1, 1024, 0, stream>>>(ws);

  int ntiles = (N + 15) >> 4;
  int blocks = (ntiles + 7) / 8;            // one tile per wave, grid-stride
  if (blocks > 2048) blocks = 2048;
  if (blocks < 1) blocks = 1;
  tm_match<<<blocks, 256, 0, stream>>>(spikes, sort_idx, match_idx, distance,
                                       ws, out, N);
}